// StationSelectionAggregator_37108517438193
// MI455X (gfx1250) — compile-verified
//
#include <hip/hip_runtime.h>
#include <hip/hip_bf16.h>

typedef __attribute__((ext_vector_type(2))) float v2f;
typedef __attribute__((ext_vector_type(8))) float v8f;

#define BB    32
#define N_ST  8
#define N_SUB 256
#define WIN   2048

// One wave32 handles one 16(w) x 16(s) tile of the output for one batch b.
//   load layout : lane = w (coalesced csi reads along contiguous w)
//   store layout: lane = s (coalesced out writes along contiguous s)
// The layout flip is done with 4x V_WMMA_F32_16X16X4_F32 against chunked
// identity selectors (bit-exact f32 transpose, no LDS).
__global__ __launch_bounds__(256) void StationSelectionAggregator_kernel(
    const float* __restrict__ schedule,  // [B, WIN, N_ST]
    const float* __restrict__ csi,       // [B, N_ST, N_SUB, WIN]
    float* __restrict__ out)             // [B, WIN, N_SUB]
{
    const int lane   = threadIdx.x & 31;
    const int waveId = blockIdx.x * (blockDim.x >> 5) + (threadIdx.x >> 5);

    const int tilesPerB = (WIN / 16) * (N_SUB / 16);   // 2048
    const int b     = waveId / tilesPerB;
    const int rem   = waveId % tilesPerB;
    const int wtile = rem % (WIN / 16);                // fast dim -> contiguous csi streaming
    const int stile = rem / (WIN / 16);

    const int half  = lane >> 4;                       // 0: K=0,1 slots   1: K=2,3 slots
    const int n     = lane & 15;                       // load: w-in-tile; store: s-in-tile
    const int w     = wtile * 16 + n;
    const int sbase = stile * 16;

    // schedule[b, w, 0..7] : 32B contiguous per lane, coalesced across half-wave
    const float4* sp = (const float4*)(schedule + ((size_t)b * WIN + w) * N_ST);
    const float4 sc0 = sp[0];
    const float4 sc1 = sp[1];
    const float sch[8] = {sc0.x, sc0.y, sc0.z, sc0.w, sc1.x, sc1.y, sc1.z, sc1.w};

    // Accumulate acc[j] = out[w, s_j],  s_j = sbase + 4*(j/2) + (j%2) + 2*half
    // (exactly the A-operand slot ordering of V_WMMA_F32_16X16X4_F32)
    const float* base = csi + (size_t)b * N_ST * N_SUB * WIN + (size_t)wtile * 16 + n;

    float acc[8];
#pragma unroll
    for (int j = 0; j < 8; ++j) acc[j] = 0.0f;

#pragma unroll
    for (int t = 0; t < N_ST; ++t) {
        const float  sw = sch[t];
        const float* bt = base + (size_t)t * N_SUB * WIN;
#pragma unroll
        for (int j = 0; j < 8; ++j) {
            const int sj = sbase + 4 * (j >> 1) + (j & 1) + 2 * half;
            acc[j] = fmaf(sw, bt[(size_t)sj * WIN], acc[j]);  // 64B coalesced per half-wave
        }
    }

    // Register-space transpose: D = sum_c A_c(16x4) * Isel_c(4x16)  == X, in D layout.
    v8f d = {};
#pragma unroll
    for (int c = 0; c < 4; ++c) {
        v2f a;
        a.x = acc[2 * c];
        a.y = acc[2 * c + 1];
        v2f bm;
        bm.x = (n == 4 * c + 0 + 2 * half) ? 1.0f : 0.0f;  // row k = 2*half
        bm.y = (n == 4 * c + 1 + 2 * half) ? 1.0f : 0.0f;  // row k = 2*half + 1
        d = __builtin_amdgcn_wmma_f32_16x16x4_f32(
                /*neg_a=*/false, a, /*neg_b=*/false, bm,
                /*c_mod=*/(short)0, d, /*reuse_a=*/false, /*reuse_b=*/false);
    }

    // D: VGPR r / lane -> out[w = 16*wtile + r + 8*half, s = sbase + n]
    float* op = out + ((size_t)b * WIN + (size_t)wtile * 16 + 8 * half) * N_SUB + sbase + n;
#pragma unroll
    for (int r = 0; r < 8; ++r) {
        op[(size_t)r * N_SUB] = d[r];                  // 64B coalesced per half-wave
    }
}

extern "C" void kernel_launch(void* const* d_in, const int* in_sizes, int n_in,
                              void* d_out, int out_size, void* d_ws, size_t ws_size,
                              hipStream_t stream) {
    (void)in_sizes; (void)n_in; (void)out_size; (void)d_ws; (void)ws_size;
    const float* schedule = (const float*)d_in[0];  // [32, 2048, 8]
    const float* csi      = (const float*)d_in[1];  // [32, 8, 256, 2048]
    float*       out      = (float*)d_out;          // [32, 2048, 256]

    const int totalWaves = BB * (WIN / 16) * (N_SUB / 16);  // 65536
    const int block = 256;                                  // 8 waves per block
    const int grid  = totalWaves / (block / 32);            // 8192
    StationSelectionAggregator_kernel<<<grid, block, 0, stream>>>(schedule, csi, out);
}